// VoiceFilterComplex_12472585027846
// MI455X (gfx1250) — compile-verified
//
#include <hip/hip_runtime.h>
#include <cmath>

// ---------------------------------------------------------------------------
// VoiceFilter (complex64) on CDNA5: all GEMMs via V_WMMA_F32_16X16X4_F32.
// Complex GEMM = 4 real WMMA accumulators (RR, II, RI, IR), combined at end.
// ---------------------------------------------------------------------------

typedef float v2f __attribute__((ext_vector_type(2)));
typedef float v8f __attribute__((ext_vector_type(8)));

constexpr int NB = 16, NC = 256, NH = 256, NT = 1000, NTREF = 500;

__device__ __forceinline__ v8f wmma4(v2f a, v2f b, v8f c) {
  // D = A(16x4 f32) * B(4x16 f32) + C(16x16 f32), wave32
  return __builtin_amdgcn_wmma_f32_16x16x4_f32(false, a, false, b, (short)0, c,
                                               false, false);
}

// complex tanh: tanh(a+ib) = [ta(1+tb^2) + i tb(1-ta^2)] / (1 + ta^2 tb^2)
__device__ __forceinline__ float2 ctanh2(float2 z) {
  float ta = tanhf(z.x);
  float tb = tanf(z.y);
  float ta2 = ta * ta, tb2 = tb * tb;
  float inv = 1.0f / (1.0f + ta2 * tb2);
  return make_float2(ta * (1.0f + tb2) * inv, tb * (1.0f - ta2) * inv);
}

// complex sigmoid: 1/(1+exp(-z)),  exp(-a-ib) = e^-a (cos b - i sin b)
__device__ __forceinline__ float2 csigmoid2(float2 z) {
  float ea = expf(-z.x);
  float sb, cb;
  sincosf(z.y, &sb, &cb);
  float wr = 1.0f + ea * cb;
  float wi = -ea * sb;
  float inv = 1.0f / (wr * wr + wi * wi);
  return make_float2(wr * inv, -wi * inv);
}

// One 16x16 complex output tile, K-loop in steps of 4 using f32 WMMA.
// loadA(m,k) -> A[m][k]  (16 x K complex), loadB(k,n) -> B[k][n] (K x 16).
template <typename LA, typename LB>
__device__ __forceinline__ void cgemm_tile(int K, LA loadA, LB loadB,
                                           v8f& outRe, v8f& outIm) {
  const int lane = threadIdx.x & 31;
  const int half = lane >> 4;   // 0: K pair {0,1}; 1: K pair {2,3}
  const int idx = lane & 15;    // A row (M) / B col (N)
  v8f rr = {0, 0, 0, 0, 0, 0, 0, 0};
  v8f ii = rr, ri = rr, ir = rr;
#pragma unroll 4
  for (int k0 = 0; k0 < K; k0 += 4) {
    const int kk = k0 + (half << 1);
    float2 a0 = loadA(idx, kk);
    float2 a1 = loadA(idx, kk + 1);
    float2 b0 = loadB(kk, idx);
    float2 b1 = loadB(kk + 1, idx);
    v2f ar = {a0.x, a1.x}, ai = {a0.y, a1.y};
    v2f br = {b0.x, b1.x}, bi = {b0.y, b1.y};
    rr = wmma4(ar, br, rr);
    ii = wmma4(ai, bi, ii);
    ri = wmma4(ar, bi, ri);
    ir = wmma4(ai, br, ir);
  }
  outRe = rr - ii;
  outIm = ri + ir;
}

// D 16x16 f32 layout: VGPR r, lanes 0-15 -> M=r, lanes 16-31 -> M=r+8; N=lane&15
template <typename ST>
__device__ __forceinline__ void store_tile(const v8f& re, const v8f& im, ST st) {
  const int lane = threadIdx.x & 31;
  const int n = lane & 15;
  const int mh = (lane >> 4) * 8;
#pragma unroll
  for (int r = 0; r < 8; ++r) st(mh + r, n, make_float2(re[r], im[r]));
}

// ---------------------------------------------------------------------------
// 1) conv: x[b,h,t] = sum_c conv_w[h,c] * mix[b,c,t] + conv_b[h]
// ---------------------------------------------------------------------------
__global__ void k_conv(const float2* __restrict__ mix,
                       const float2* __restrict__ W,
                       const float2* __restrict__ bias,
                       float2* __restrict__ x) {
  const int tt = blockIdx.x * 16;  // N = t
  const int hh = blockIdx.y * 16;  // M = h
  const int b = blockIdx.z;
  const float2* Mb = mix + (size_t)b * NC * NT;
  __builtin_prefetch(Mb + tt, 0, 0);  // global_prefetch_b8
  v8f re, im;
  cgemm_tile(
      NC, [&](int m, int k) { return W[(size_t)(hh + m) * NC + k]; },
      [&](int k, int n) {
        int t = tt + n;
        if (t >= NT) t = NT - 1;  // clamp; dead cols never stored
        return Mb[(size_t)k * NT + t];
      },
      re, im);
  store_tile(re, im, [&](int m, int n, float2 v) {
    int t = tt + n;
    if (t >= NT) return;
    float2 bv = bias[hh + m];
    v.x += bv.x;
    v.y += bv.y;
    x[((size_t)b * NH + hh + m) * NT + t] = v;
  });
}

// ---------------------------------------------------------------------------
// 2) u0[b,t,n] = sum_c emb_Wih0[n,c] * ref[b,c,t] + bih0[n] + bhh0[n]
//    rows flat over (b,t): M = B*TREF = 8000 (500 tiles exactly)
// ---------------------------------------------------------------------------
__global__ void k_u0(const float2* __restrict__ ref,
                     const float2* __restrict__ Wih,
                     const float2* __restrict__ bih,
                     const float2* __restrict__ bhh,
                     float2* __restrict__ u0) {
  const int nn = blockIdx.x * 16;
  const int mm = blockIdx.y * 16;
  v8f re, im;
  cgemm_tile(
      NC,
      [&](int m, int k) {
        int flat = mm + m;
        int b = flat / NTREF, t = flat % NTREF;
        return ref[((size_t)b * NC + k) * NTREF + t];
      },
      [&](int k, int n) { return Wih[(size_t)(nn + n) * NC + k]; }, re, im);
  store_tile(re, im, [&](int m, int n, float2 v) {
    float2 c1 = bih[nn + n], c2 = bhh[nn + n];
    v.x += c1.x + c2.x;
    v.y += c1.y + c2.y;
    u0[(size_t)(mm + m) * NH + nn + n] = v;
  });
}

// ---------------------------------------------------------------------------
// 3) u1[b,t,n] = sum_k emb_Wih1[n,k] * h1[b,t,k] + bih1[n] + bhh1[n]
// ---------------------------------------------------------------------------
__global__ void k_u1(const float2* __restrict__ h1,
                     const float2* __restrict__ Wih,
                     const float2* __restrict__ bih,
                     const float2* __restrict__ bhh,
                     float2* __restrict__ u1) {
  const int nn = blockIdx.x * 16;
  const int mm = blockIdx.y * 16;
  v8f re, im;
  cgemm_tile(
      NH, [&](int m, int k) { return h1[(size_t)(mm + m) * NH + k]; },
      [&](int k, int n) { return Wih[(size_t)(nn + n) * NH + k]; }, re, im);
  store_tile(re, im, [&](int m, int n, float2 v) {
    float2 c1 = bih[nn + n], c2 = bhh[nn + n];
    v.x += c1.x + c2.x;
    v.y += c1.y + c2.y;
    u1[(size_t)(mm + m) * NH + nn + n] = v;
  });
}

// ---------------------------------------------------------------------------
// 4) Persistent RNN scan body. One workgroup, 16 waves, state h (16x256 cplx)
//    in LDS. Per step: h = ctanh(u[t] + h @ Whh^T); wave owns one N-tile.
// ---------------------------------------------------------------------------
__device__ __forceinline__ void scan_body(const float2* __restrict__ u,
                                          const float2* __restrict__ Whh,
                                          float2* __restrict__ yseq,
                                          float2* __restrict__ dvec, int Ts,
                                          int reverse) {
  __shared__ float2 hs[16][NH];  // 32 KB of the 320 KB LDS
  const int lane = threadIdx.x & 31;
  const int wave = threadIdx.x >> 5;  // 0..15
  const int nn = wave * 16;
  const int half = lane >> 4;
  const int idx = lane & 15;

  for (int i = threadIdx.x; i < 16 * NH; i += 512)
    (&hs[0][0])[i] = make_float2(0.f, 0.f);
  __syncthreads();

  const float2* wrow = Whh + (size_t)(nn + idx) * NH;  // B[k][n]=Whh[n][k]

  for (int s = 0; s < Ts; ++s) {
    const int t = reverse ? (Ts - 1 - s) : s;
    // prefetch next step's epilogue operand while WMMAs run
    if (s + 1 < Ts) {
      const int tn = reverse ? (t - 1) : (t + 1);
      __builtin_prefetch(&u[((size_t)(half * 8) * Ts + tn) * NH + nn + idx], 0,
                         0);
    }
    v8f rr = {0, 0, 0, 0, 0, 0, 0, 0};
    v8f ii = rr, ri = rr, ir = rr;
#pragma unroll 4
    for (int k0 = 0; k0 < NH; k0 += 4) {
      const int kk = k0 + (half << 1);
      float2 a0 = hs[idx][kk];
      float2 a1 = hs[idx][kk + 1];
      float2 b0 = wrow[kk];
      float2 b1 = wrow[kk + 1];
      v2f ar = {a0.x, a1.x}, ai = {a0.y, a1.y};
      v2f br = {b0.x, b1.x}, bi = {b0.y, b1.y};
      rr = wmma4(ar, br, rr);
      ii = wmma4(ai, bi, ii);
      ri = wmma4(ar, bi, ri);
      ir = wmma4(ai, br, ir);
    }
    v8f re = rr - ii, im = ri + ir;
    __syncthreads();  // everyone done reading hs
#pragma unroll
    for (int r = 0; r < 8; ++r) {
      const int m = half * 8 + r;  // batch row
      const int n = nn + idx;
      float2 uv = u[((size_t)m * Ts + t) * NH + n];
      float2 hv = ctanh2(make_float2(re[r] + uv.x, im[r] + uv.y));
      hs[m][n] = hv;
      if (yseq) yseq[((size_t)m * Ts + t) * NH + n] = hv;
    }
    __syncthreads();
  }
  if (dvec) {
#pragma unroll
    for (int r = 0; r < 8; ++r) {
      const int m = half * 8 + r;
      const int n = nn + idx;
      dvec[(size_t)m * NH + n] = hs[m][n];
    }
  }
}

__global__ void __launch_bounds__(512) k_scan(const float2* __restrict__ u,
                                              const float2* __restrict__ Whh,
                                              float2* __restrict__ yseq,
                                              float2* __restrict__ dvec,
                                              int Ts, int reverse) {
  scan_body(u, Whh, yseq, dvec, Ts, reverse);
}

// Both BiRNN directions in ONE launch: block 0 = forward, block 1 = backward.
// Each block runs on its own WGP with its own LDS state -> the two 1000-step
// latency chains overlap instead of serializing on the stream.
__global__ void __launch_bounds__(512) k_scan_bi(
    const float2* __restrict__ uf, const float2* __restrict__ Whhf,
    float2* __restrict__ yf, const float2* __restrict__ ub,
    const float2* __restrict__ Whhb, float2* __restrict__ yb, int Ts) {
  if (blockIdx.x == 0) {
    scan_body(uf, Whhf, yf, nullptr, Ts, 0);
  } else {
    scan_body(ub, Whhb, yb, nullptr, Ts, 1);
  }
}

// ---------------------------------------------------------------------------
// 5) v[dir][b,n] = sum_k dvec[b,k] * Wih_dir[n, H+k] + bih+bhh  (t-invariant)
// ---------------------------------------------------------------------------
__global__ void k_vdvec(const float2* __restrict__ dvec,
                        const float2* __restrict__ Wf,
                        const float2* __restrict__ bihf,
                        const float2* __restrict__ bhhf,
                        const float2* __restrict__ Wb,
                        const float2* __restrict__ bihb,
                        const float2* __restrict__ bhhb,
                        float2* __restrict__ v) {
  const int nn = blockIdx.x * 16;
  const int dir = blockIdx.y;
  const float2* Wih = dir ? Wb : Wf;
  const float2* c1p = dir ? bihb : bihf;
  const float2* c2p = dir ? bhhb : bhhf;
  v8f re, im;
  cgemm_tile(
      NH, [&](int m, int k) { return dvec[(size_t)m * NH + k]; },
      [&](int k, int n) { return Wih[(size_t)(nn + n) * (2 * NH) + NH + k]; },
      re, im);
  store_tile(re, im, [&](int m, int n, float2 val) {
    float2 c1 = c1p[nn + n], c2 = c2p[nn + n];
    val.x += c1.x + c2.x;
    val.y += c1.y + c2.y;
    v[((size_t)dir * NB + m) * NH + nn + n] = val;
  });
}

// ---------------------------------------------------------------------------
// 6) u_{f,b}[b,t,n] = sum_{k<H} Wih[n,k] * x[b,k,t] + v[dir][b,n]
// ---------------------------------------------------------------------------
__global__ void k_umain(const float2* __restrict__ x,
                        const float2* __restrict__ Wf,
                        const float2* __restrict__ Wb,
                        const float2* __restrict__ v,
                        float2* __restrict__ uf, float2* __restrict__ ub) {
  const int tt = blockIdx.x * 16;  // M = t
  const int nn = blockIdx.y * 16;  // N = n
  const int bz = blockIdx.z;       // b*2 + dir
  const int b = bz >> 1, dir = bz & 1;
  const float2* Wih = dir ? Wb : Wf;
  float2* u = dir ? ub : uf;
  v8f re, im;
  cgemm_tile(
      NH,
      [&](int m, int k) {
        int t = tt + m;
        if (t >= NT) t = NT - 1;  // clamp; dead rows never stored
        return x[((size_t)b * NH + k) * NT + t];
      },
      [&](int k, int n) { return Wih[(size_t)(nn + n) * (2 * NH) + k]; }, re,
      im);
  store_tile(re, im, [&](int m, int n, float2 val) {
    int t = tt + m;
    if (t >= NT) return;
    float2 vv = v[((size_t)dir * NB + b) * NH + nn + n];
    val.x += vv.x;
    val.y += vv.y;
    u[((size_t)b * NT + t) * NH + nn + n] = val;
  });
}

// ---------------------------------------------------------------------------
// 7) fc1: y1[m,n] = sum_{k<2H} cat(yf,yb)[m,k] * fc1_W[n,k] + fc1_b[n]
// ---------------------------------------------------------------------------
__global__ void k_fc1(const float2* __restrict__ yf,
                      const float2* __restrict__ yb,
                      const float2* __restrict__ W,
                      const float2* __restrict__ bias,
                      float2* __restrict__ y1) {
  const int nn = blockIdx.x * 16;
  const int mm = blockIdx.y * 16;
  v8f re, im;
  cgemm_tile(
      2 * NH,
      [&](int m, int k) {
        size_t row = mm + m;
        return (k < NH) ? yf[row * NH + k] : yb[row * NH + (k - NH)];
      },
      [&](int k, int n) { return W[(size_t)(nn + n) * (2 * NH) + k]; }, re, im);
  store_tile(re, im, [&](int m, int n, float2 v) {
    float2 c = bias[nn + n];
    v.x += c.x;
    v.y += c.y;
    y1[(size_t)(mm + m) * NH + nn + n] = v;
  });
}

// ---------------------------------------------------------------------------
// 8) fc2 + complex sigmoid + mask*mix, fused with (B,T,C)->(B,C,T) transpose
// ---------------------------------------------------------------------------
__global__ void k_fc2(const float2* __restrict__ y1,
                      const float2* __restrict__ W,
                      const float2* __restrict__ bias,
                      const float2* __restrict__ mix,
                      float2* __restrict__ maskOut,
                      float2* __restrict__ predOut) {
  const int nn = blockIdx.x * 16;  // c
  const int mm = blockIdx.y * 16;  // flat (b,t)
  v8f re, im;
  cgemm_tile(
      NH, [&](int m, int k) { return y1[(size_t)(mm + m) * NH + k]; },
      [&](int k, int n) { return W[(size_t)(nn + n) * NH + k]; }, re, im);
  store_tile(re, im, [&](int m, int n, float2 val) {
    int flat = mm + m;
    int b = flat / NT, t = flat % NT;
    int c = nn + n;
    float2 bv = bias[c];
    float2 mk = csigmoid2(make_float2(val.x + bv.x, val.y + bv.y));
    size_t oi = ((size_t)b * NC + c) * NT + t;
    maskOut[oi] = mk;
    float2 mx = mix[oi];
    predOut[oi] =
        make_float2(mk.x * mx.x - mk.y * mx.y, mk.x * mx.y + mk.y * mx.x);
  });
}

// ---------------------------------------------------------------------------
// 9) dvec_rep[b,h,t] = dvec[b,h]
// ---------------------------------------------------------------------------
__global__ void k_dvecrep(const float2* __restrict__ dvec,
                          float2* __restrict__ out) {
  size_t i = (size_t)blockIdx.x * blockDim.x + threadIdx.x;
  if (i >= (size_t)NB * NH * NT) return;
  out[i] = dvec[i / NT];
}

// ---------------------------------------------------------------------------
extern "C" void kernel_launch(void* const* d_in, const int* in_sizes, int n_in,
                              void* d_out, int out_size, void* d_ws,
                              size_t ws_size, hipStream_t stream) {
  (void)in_sizes; (void)n_in; (void)out_size; (void)ws_size;

  const float2* mix   = (const float2*)d_in[0];
  const float2* ref   = (const float2*)d_in[1];
  const float2* convw = (const float2*)d_in[2];
  const float2* convb = (const float2*)d_in[3];
  const float2* eWih0 = (const float2*)d_in[4];
  const float2* eWhh0 = (const float2*)d_in[5];
  const float2* ebih0 = (const float2*)d_in[6];
  const float2* ebhh0 = (const float2*)d_in[7];
  const float2* eWih1 = (const float2*)d_in[8];
  const float2* eWhh1 = (const float2*)d_in[9];
  const float2* ebih1 = (const float2*)d_in[10];
  const float2* ebhh1 = (const float2*)d_in[11];
  const float2* rWihF = (const float2*)d_in[12];
  const float2* rWhhF = (const float2*)d_in[13];
  const float2* rbihF = (const float2*)d_in[14];
  const float2* rbhhF = (const float2*)d_in[15];
  const float2* rWihB = (const float2*)d_in[16];
  const float2* rWhhB = (const float2*)d_in[17];
  const float2* rbihB = (const float2*)d_in[18];
  const float2* rbhhB = (const float2*)d_in[19];
  const float2* fc1W  = (const float2*)d_in[20];
  const float2* fc1b  = (const float2*)d_in[21];
  const float2* fc2W  = (const float2*)d_in[22];
  const float2* fc2b  = (const float2*)d_in[23];

  const size_t N1 = (size_t)NB * NC * NT;    // 4,096,000 complex
  const size_t N2 = (size_t)NB * NC * NTREF; // 2,048,000 complex

  float2* ws   = (float2*)d_ws;
  float2* x    = ws;                 // (B,H,T)
  float2* u0   = x + N1;             // (B,TREF,H)  (reused as u1)
  float2* h1   = u0 + N2;            // (B,TREF,H)
  float2* dvec = h1 + N2;            // (B,H)
  float2* v    = dvec + NB * NH;     // (2,B,H)
  float2* uf   = v + 2 * NB * NH;    // (B,T,H)     (reused as y1)
  float2* ub   = uf + N1;            // (B,T,H)
  float2* yf   = ub + N1;            // (B,T,H)
  float2* yb   = yf + N1;            // (B,T,H)

  float2* pred  = (float2*)d_out;    // (B,C,T)
  float2* maskO = pred + N1;         // (B,C,T)
  float2* dvrep = pred + 2 * N1;     // (B,H,T)

  // 1) conv einsum
  k_conv<<<dim3((NT + 15) / 16, NH / 16, NB), 32, 0, stream>>>(mix, convw,
                                                               convb, x);
  // 2) embedding layer 0: input projection then scan
  k_u0<<<dim3(NH / 16, (NB * NTREF) / 16), 32, 0, stream>>>(ref, eWih0, ebih0,
                                                            ebhh0, u0);
  k_scan<<<1, 512, 0, stream>>>(u0, eWhh0, h1, nullptr, NTREF, 0);
  // 3) embedding layer 1 (u1 aliases u0; only final state needed)
  k_u1<<<dim3(NH / 16, (NB * NTREF) / 16), 32, 0, stream>>>(h1, eWih1, ebih1,
                                                            ebhh1, u0);
  k_scan<<<1, 512, 0, stream>>>(u0, eWhh1, nullptr, dvec, NTREF, 0);
  // 4) t-invariant dvec projection for both directions
  k_vdvec<<<dim3(NH / 16, 2), 32, 0, stream>>>(dvec, rWihF, rbihF, rbhhF,
                                               rWihB, rbihB, rbhhB, v);
  // 5) BiRNN input projections
  k_umain<<<dim3((NT + 15) / 16, NH / 16, NB * 2), 32, 0, stream>>>(
      x, rWihF, rWihB, v, uf, ub);
  // 6) forward + backward scans concurrently (2 blocks, 2 WGPs)
  k_scan_bi<<<2, 512, 0, stream>>>(uf, rWhhF, yf, ub, rWhhB, yb, NT);
  // 7) fc1 (y1 aliases uf, free after scans)
  k_fc1<<<dim3(NH / 16, (NB * NT) / 16), 32, 0, stream>>>(yf, yb, fc1W, fc1b,
                                                          uf);
  // 8) fc2 + sigmoid + mask*mix (+ transpose) -> pred, mask
  k_fc2<<<dim3(NC / 16, (NB * NT) / 16), 32, 0, stream>>>(uf, fc2W, fc2b, mix,
                                                          maskO, pred);
  // 9) dvec_rep broadcast
  {
    size_t n = (size_t)NB * NH * NT;
    k_dvecrep<<<dim3((unsigned)((n + 255) / 256)), 256, 0, stream>>>(dvec,
                                                                     dvrep);
  }
}